// BCEDiceLoss_blobPunish_6047313953507
// MI455X (gfx1250) — compile-verified
//
#include <hip/hip_runtime.h>
#include <cstdint>

#define BATCH 16
#define H 512
#define W 512
#define IMG (H * W)
#define NPIX (BATCH * IMG)          // 4194304
#define TILE 64
#define RAD 8
#define PADW (TILE + 2 * RAD)       // 80
#define PCELLS (PADW * PADW)        // 6400
#define NROUNDS 25                  // 25 * RAD = 200 iterations
#define BITWORDS 131073             // ceil((NPIX+1)/32)
#define BITWORDS_PAD 131104

// LDS tile layout: packed 80-pitch (TDM writes contiguously; pad_interval is
// powers-of-2 only so a wider pitch is not expressible). One zero row above
// and below (LOFF shift) makes all vertical neighbor reads in-bounds;
// horizontal wrap at tile edges only corrupts the outermost halo ring, which
// is stale by construction (8-deep halo, 8 local iterations) and never
// reaches the written-back interior.
#define LOFF (PADW + 1)             // 81
#define LSIZE (PCELLS + 2 * LOFF)   // 6562 floats

typedef unsigned int v4u __attribute__((ext_vector_type(4)));
typedef int v8i __attribute__((ext_vector_type(8)));
typedef int v4i __attribute__((ext_vector_type(4)));

// ---------------------------------------------------------------------------
// TDM 2D tile load: 80x80 f32 tile from a W-stride tensor into LDS.
// D# layout per CDNA5 ISA (group0 128b, group1 256b).
// ---------------------------------------------------------------------------
__device__ __forceinline__ void tdm_load_tile_f32(unsigned lds_off, const float* gsrc) {
  unsigned long long ga = (unsigned long long)(uintptr_t)gsrc;
  v4u g0;
  g0[0] = 1u;                                            // count=1, is_restore=0, gather off
  g0[1] = lds_off;                                       // lds_addr (bytes)
  g0[2] = (unsigned)(ga & 0xFFFFFFFFull);                // global_addr[31:0]
  g0[3] = (unsigned)((ga >> 32) & 0x1FFFFFFull)          // global_addr[56:32]
        | (2u << 30);                                    // type=2 ("image")
  v8i g1;
  g1[0] = (2 << 16);                                     // data_size=2 (4 bytes)
  g1[1] = (int)((unsigned)W << 16);                      // tensor_dim0[15:0] @ [63:48]
  g1[2] = (int)((unsigned)H << 16);                      // dim0 hi16=0 | tensor_dim1[15:0]
  g1[3] = (int)((unsigned)PADW << 16);                   // dim1 hi16=0 | tile_dim0
  g1[4] = PADW;                                          // tile_dim1 | tile_dim2=0
  g1[5] = W;                                             // tensor_dim0_stride[31:0]
  g1[6] = 0;                                             // stride hi | dim1_stride lo
  g1[7] = 0;
  v4i z4 = {0, 0, 0, 0};
#if __clang_major__ >= 23
  v8i z8 = {0, 0, 0, 0, 0, 0, 0, 0};
  __builtin_amdgcn_tensor_load_to_lds(g0, g1, z4, z4, z8, 0);
#else
  __builtin_amdgcn_tensor_load_to_lds(g0, g1, z4, z4, 0);
#endif
}

// ---------------------------------------------------------------------------
// Global max (stage 1): fixed-tree block reduction -> one partial per block.
// ---------------------------------------------------------------------------
__global__ void __launch_bounds__(256) partial_max_kernel(const float* __restrict__ x,
                                                          float* __restrict__ part) {
  __shared__ float red[256];
  float m = -3.402823466e38f;
  for (int i = blockIdx.x * 256 + threadIdx.x; i < NPIX; i += gridDim.x * 256)
    m = fmaxf(m, x[i]);
  red[threadIdx.x] = m;
  __syncthreads();
  for (int s = 128; s > 0; s >>= 1) {
    if (threadIdx.x < s) red[threadIdx.x] = fmaxf(red[threadIdx.x], red[threadIdx.x + s]);
    __syncthreads();
  }
  if (threadIdx.x == 0) part[blockIdx.x] = red[0];
}

__global__ void final_max_kernel(const float* __restrict__ part, int n, float* __restrict__ out) {
  if (threadIdx.x == 0 && blockIdx.x == 0) {
    float m = -3.402823466e38f;
    for (int i = 0; i < n; ++i) m = fmaxf(m, part[i]);
    out[0] = m;
  }
}

// mask = x > max/2 ; labels = mask * (flat_index + 1)
__global__ void __launch_bounds__(256) init_kernel(const float* __restrict__ x,
                                                   const float* __restrict__ maxv,
                                                   float* __restrict__ mask,
                                                   float* __restrict__ lab) {
  const float th = maxv[0] * 0.5f;
  for (int i = blockIdx.x * 256 + threadIdx.x; i < NPIX; i += gridDim.x * 256) {
    float m = x[i] > th ? 1.0f : 0.0f;
    mask[i] = m;
    lab[i] = m * (float)(i + 1);   // ids <= 2^22, exact in f32
  }
}

__global__ void clear_kernel(unsigned* __restrict__ p, int n) {
  for (int i = blockIdx.x * 256 + threadIdx.x; i < n; i += gridDim.x * 256) p[i] = 0u;
}

// ---------------------------------------------------------------------------
// 8 iterations of masked 3x3 max-pool on a 64x64 tile with 8-deep halo, all in
// LDS. Interior tiles DMA their 80x80 input tile via the Tensor Data Mover
// (TENSORcnt); boundary tiles use guarded loads (zero pad == reduce_window
// 'SAME' with init 0, since labels are nonnegative).
// Each thread owns a 5x5 cell block (16x16 blocks = 256 threads): separable
// 3x3 max via 7 column loads (49 LDS reads) + 70 vertical + 50 horizontal
// v_max per 25 cells, all with constant offsets from one base.
// ---------------------------------------------------------------------------
__global__ void __launch_bounds__(256)
pool_tile_kernel(const float* __restrict__ labSrc, float* __restrict__ labDst,
                 const float* __restrict__ mask) {
  __shared__ float slab[LSIZE];
  __shared__ float smsk[LSIZE];
  const int tx = blockIdx.x, ty = blockIdx.y, img = blockIdx.z;
  const int x0 = tx * TILE - RAD, y0 = ty * TILE - RAD;
  const float* labImg = labSrc + (size_t)img * IMG;
  const float* mskImg = mask + (size_t)img * IMG;
  const int tid = threadIdx.x;
  const bool interior = (x0 >= 0) && (y0 >= 0) && (x0 + PADW <= W) && (y0 + PADW <= H);

  // Zero the top/bottom guard regions: [0, LOFF) and [LOFF+PCELLS, LSIZE).
  if (tid < LOFF) {
    slab[tid] = 0.0f;
    smsk[tid] = 0.0f;
    slab[LOFF + PCELLS + tid] = 0.0f;
    smsk[LOFF + PCELLS + tid] = 0.0f;
  }

  if (interior) {
    if ((tid >> 5) == 0) {  // wave 0 issues the two tile DMAs
      tdm_load_tile_f32((unsigned)(uintptr_t)&slab[LOFF], labImg + y0 * W + x0);
      tdm_load_tile_f32((unsigned)(uintptr_t)&smsk[LOFF], mskImg + y0 * W + x0);
      __builtin_amdgcn_s_wait_tensorcnt((short)0);
    }
    __syncthreads();
  } else {
    for (int idx = tid; idx < PCELLS; idx += 256) {
      int lx = idx % PADW, ly = idx / PADW;
      int gx = x0 + lx, gy = y0 + ly;
      bool ok = (gx >= 0) && (gx < W) && (gy >= 0) && (gy < H);
      slab[LOFF + idx] = ok ? labImg[gy * W + gx] : 0.0f;
      smsk[LOFF + idx] = ok ? mskImg[gy * W + gx] : 0.0f;
    }
    __syncthreads();
  }

  // This thread's 5x5 block of cells.
  const int bx = (tid & 15) * 5;
  const int by = (tid >> 4) * 5;
  const int base = LOFF + by * PADW + bx;   // address of cell (bx, by)

  // Hoist the 25 mask values into registers.
  float mreg[25];
#pragma unroll
  for (int r = 0; r < 5; ++r)
#pragma unroll
    for (int c = 0; c < 5; ++c) mreg[r * 5 + c] = smsk[base + r * PADW + c];

  float nv[25];
  for (int it = 0; it < RAD; ++it) {
    float vm[35];  // vertical 3-max per (column 0..6, row 0..4)
#pragma unroll
    for (int c = 0; c < 7; ++c) {
      const int cb = base + (c - 1);
      float v0 = slab[cb - PADW];
      float v1 = slab[cb];
      float v2 = slab[cb + PADW];
      float v3 = slab[cb + 2 * PADW];
      float v4 = slab[cb + 3 * PADW];
      float v5 = slab[cb + 4 * PADW];
      float v6 = slab[cb + 5 * PADW];
      vm[c * 5 + 0] = fmaxf(fmaxf(v0, v1), v2);
      vm[c * 5 + 1] = fmaxf(fmaxf(v1, v2), v3);
      vm[c * 5 + 2] = fmaxf(fmaxf(v2, v3), v4);
      vm[c * 5 + 3] = fmaxf(fmaxf(v3, v4), v5);
      vm[c * 5 + 4] = fmaxf(fmaxf(v4, v5), v6);
    }
#pragma unroll
    for (int r = 0; r < 5; ++r)
#pragma unroll
      for (int c = 0; c < 5; ++c)
        nv[r * 5 + c] =
            fmaxf(fmaxf(vm[c * 5 + r], vm[(c + 1) * 5 + r]), vm[(c + 2) * 5 + r]) *
            mreg[r * 5 + c];
    __syncthreads();
#pragma unroll
    for (int r = 0; r < 5; ++r)
#pragma unroll
      for (int c = 0; c < 5; ++c) slab[base + r * PADW + c] = nv[r * 5 + c];
    __syncthreads();
  }

  float* dstImg = labDst + (size_t)img * IMG;
  for (int idx = tid; idx < TILE * TILE; idx += 256) {
    int lx = idx % TILE, ly = idx / TILE;
    dstImg[(ty * TILE + ly) * W + (tx * TILE + lx)] =
        slab[LOFF + (ly + RAD) * PADW + (lx + RAD)];
  }
}

// Count-distinct: set bit per label id, then popcount (integer atomics = deterministic).
__global__ void __launch_bounds__(256) bitmap_kernel(const float* __restrict__ lab,
                                                     unsigned* __restrict__ bm) {
  for (int i = blockIdx.x * 256 + threadIdx.x; i < NPIX; i += gridDim.x * 256) {
    int id = (int)lab[i];
    atomicOr(&bm[id >> 5], 1u << (id & 31));
  }
}

__global__ void __launch_bounds__(256) popcount_kernel(const unsigned* __restrict__ bm, int n,
                                                       unsigned* __restrict__ cnt) {
  __shared__ unsigned red[256];
  unsigned s = 0;
  for (int i = blockIdx.x * 256 + threadIdx.x; i < n; i += gridDim.x * 256) s += __popc(bm[i]);
  red[threadIdx.x] = s;
  __syncthreads();
  for (int st = 128; st > 0; st >>= 1) {
    if (threadIdx.x < st) red[threadIdx.x] += red[threadIdx.x + st];
    __syncthreads();
  }
  if (threadIdx.x == 0) atomicAdd(cnt, red[0]);
}

// BCE + per-image dice partial sums: 64 blocks/image, fixed-tree reduction.
__global__ void __launch_bounds__(256) bce_partial_kernel(const float* __restrict__ in,
                                                          const float* __restrict__ tg,
                                                          float* __restrict__ part) {
  __shared__ float red[4][256];
  const int img = blockIdx.x >> 6, chunk = blockIdx.x & 63;
  const float* x = in + (size_t)img * IMG + (size_t)chunk * 4096;
  const float* t = tg + (size_t)img * IMG + (size_t)chunk * 4096;
  float sb = 0.f, spt = 0.f, sp = 0.f, st = 0.f;
#pragma unroll
  for (int k = 0; k < 16; ++k) {
    int i = threadIdx.x + k * 256;
    float xi = x[i], ti = t[i];
    sb += fmaxf(xi, 0.f) - xi * ti + log1pf(expf(-fabsf(xi)));
    float p = 1.f / (1.f + expf(-xi));
    spt += p * ti;
    sp += p;
    st += ti;
  }
  red[0][threadIdx.x] = sb;
  red[1][threadIdx.x] = spt;
  red[2][threadIdx.x] = sp;
  red[3][threadIdx.x] = st;
  __syncthreads();
  for (int s = 128; s > 0; s >>= 1) {
    if (threadIdx.x < s)
      for (int j = 0; j < 4; ++j) red[j][threadIdx.x] += red[j][threadIdx.x + s];
    __syncthreads();
  }
  if (threadIdx.x == 0)
    for (int j = 0; j < 4; ++j) part[blockIdx.x * 4 + j] = red[j][0];
}

__global__ void final_kernel(const float* __restrict__ part,
                             const unsigned* __restrict__ cntIn,
                             const unsigned* __restrict__ cntTg,
                             float* __restrict__ out) {
  if (threadIdx.x != 0 || blockIdx.x != 0) return;
  float bce = 0.f;
  for (int b = 0; b < BATCH * 64; ++b) bce += part[b * 4];
  bce /= (float)NPIX;
  float dsum = 0.f;
  for (int img = 0; img < BATCH; ++img) {
    float pt = 0.f, ps = 0.f, ts = 0.f;
    for (int c = 0; c < 64; ++c) {
      int r = (img * 64 + c) * 4;
      pt += part[r + 1];
      ps += part[r + 2];
      ts += part[r + 3];
    }
    dsum += (2.f * pt + 1e-5f) / (ps + ts + 1e-5f);
  }
  float dice = 1.f - dsum / (float)BATCH;
  float nl = (float)(*cntIn) - 1.f;   // background id 0 counted -> subtract 1
  float nt = (float)(*cntTg);         // reference does NOT subtract 1 here
  float pen = sqrtf(nl / nt);
  if (isinf(pen) || isnan(pen)) pen = (float)BATCH;
  pen = fminf(fmaxf(pen, 1.f), (float)BATCH);
  out[0] = 0.5f * (bce + dice) + pen;
}

// ---------------------------------------------------------------------------
extern "C" void kernel_launch(void* const* d_in, const int* in_sizes, int n_in,
                              void* d_out, int out_size, void* d_ws, size_t ws_size,
                              hipStream_t stream) {
  const float* in = (const float*)d_in[0];
  const float* tg = (const float*)d_in[1];

  float* wsf = (float*)d_ws;
  float* mask = wsf;                           // NPIX
  float* labA = wsf + (size_t)NPIX;            // NPIX
  float* labB = wsf + 2 * (size_t)NPIX;        // NPIX
  float* partMax = wsf + 3 * (size_t)NPIX;     // 2048
  float* maxV = partMax + 2048;                // 1
  float* bceP = maxV + 1;                      // 4096
  unsigned* bitmap = (unsigned*)(bceP + 4096); // BITWORDS_PAD
  unsigned* cnts = bitmap + BITWORDS_PAD;      // 2

  const float* srcs[2] = {in, tg};
  for (int ti = 0; ti < 2; ++ti) {
    const float* x = srcs[ti];
    partial_max_kernel<<<2048, 256, 0, stream>>>(x, partMax);
    final_max_kernel<<<1, 32, 0, stream>>>(partMax, 2048, maxV);
    init_kernel<<<2048, 256, 0, stream>>>(x, maxV, mask, labA);
    clear_kernel<<<512, 256, 0, stream>>>(bitmap, BITWORDS_PAD);
    clear_kernel<<<1, 32, 0, stream>>>(cnts + ti, 1);
    float* s = labA;
    float* d = labB;
    for (int r = 0; r < NROUNDS; ++r) {  // 25 rounds x 8 LDS iterations = 200
      pool_tile_kernel<<<dim3(W / TILE, H / TILE, BATCH), 256, 0, stream>>>(s, d, mask);
      float* tmp = s; s = d; d = tmp;
    }
    bitmap_kernel<<<2048, 256, 0, stream>>>(s, bitmap);
    popcount_kernel<<<256, 256, 0, stream>>>(bitmap, BITWORDS, cnts + ti);
  }
  bce_partial_kernel<<<BATCH * 64, 256, 0, stream>>>(in, tg, bceP);
  final_kernel<<<1, 32, 0, stream>>>(bceP, cnts + 0, cnts + 1, (float*)d_out);
}